// GanaSAGEConv_27522150433359
// MI455X (gfx1250) — compile-verified
//
#include <hip/hip_runtime.h>
#include <math.h>

typedef __attribute__((ext_vector_type(2))) float v2f;
typedef __attribute__((ext_vector_type(8))) float v8f;

// ---------------------------------------------------------------------------
// Native device-scope f32 atomic add (non-returning -> STOREcnt, RMW at L2).
// Pinned via inline asm so the backend cannot expand it into a CAS loop.
// ---------------------------------------------------------------------------
__device__ __forceinline__ void atomic_add_f32_dev(float* addr, float val) {
    asm volatile("global_atomic_add_f32 %0, %1, off scope:SCOPE_DEV"
                 :
                 : "v"(addr), "v"(val)
                 : "memory");
}

// ---------------------------------------------------------------------------
// Zero fill (float4 granularity)
// ---------------------------------------------------------------------------
__global__ void fill_zero_f4(float* __restrict__ p, long n4) {
    long t = (long)blockIdx.x * blockDim.x + threadIdx.x;
    if (t < n4) {
        ((float4*)p)[t] = make_float4(0.f, 0.f, 0.f, 0.f);
    }
}

// ---------------------------------------------------------------------------
// Edge scatter: one wave (32 lanes) per edge, lane l moves features 4l..4l+3.
// agg[dst] += x[src]; cnt[dst] += 1.  F is fixed at 128 (all layer inputs).
// Lanes read consecutive 16B chunks of x[src] (coalesced) and issue
// global_atomic_add_f32 to consecutive addresses of agg[dst].
// ---------------------------------------------------------------------------
__global__ void sage_scatter(const float* __restrict__ x,
                             const int* __restrict__ ei,
                             float* __restrict__ agg,
                             float* __restrict__ cnt,
                             long E) {
    long t = (long)blockIdx.x * blockDim.x + threadIdx.x;
    long e = t >> 5;
    if (e >= E) return;
    int  l   = (int)(t & 31);
    long src = (long)ei[e];
    long dst = (long)ei[E + e];
    const float4 v = *(const float4*)(x + src * 128 + l * 4);
    float* a = agg + dst * 128 + l * 4;
    atomic_add_f32_dev(a + 0, v.x);
    atomic_add_f32_dev(a + 1, v.y);
    atomic_add_f32_dev(a + 2, v.z);
    atomic_add_f32_dev(a + 3, v.w);
    if (l == 0) {
        atomic_add_f32_dev(cnt + dst, 1.0f);
    }
}

// ---------------------------------------------------------------------------
// Fused SAGE dense part, f32 WMMA 16x16x4:
//   out[n,h] = sum_k (agg[n,k]/max(cnt[n],1))*Wl[h,k] + sum_k xin[n,k]*Wr[h,k] + b[h]
// One wave32 computes one 16(M=nodes) x 16(N=out features) tile.
// A 16x4 f32 frag: lane L -> M=L&15, a.x=K(klo), a.y=K(klo+1), klo=k+2*(L>>4)
// B  4x16 f32 frag: lane L -> N=L&15, b.x=B[klo][N]=W[N][klo], b.y=W[N][klo+1]
// C/D 16x16: VGPR v, lane L -> M = v + 8*(L>>4), N = L&15
// ---------------------------------------------------------------------------
__global__ void sage_gemm_wmma(const float* __restrict__ agg,
                               const float* __restrict__ cnt,
                               const float* __restrict__ xin,
                               const float* __restrict__ Wl,
                               const float* __restrict__ Wr,
                               const float* __restrict__ bias,
                               float* __restrict__ out,
                               int n_nodes, int Fin, int Hout, int do_relu) {
    int wave = (int)(((long)blockIdx.x * blockDim.x + threadIdx.x) >> 5);
    int lane = threadIdx.x & 31;
    int ntiles_h = Hout >> 4;
    int total    = (n_nodes >> 4) * ntiles_h;
    if (wave >= total) return;          // wave-uniform: EXEC all-ones below

    int tile_m = wave / ntiles_h;
    int tile_n = wave % ntiles_h;
    int row  = lane & 15;
    int hi   = lane >> 4;               // selects K pair (k, k+1) vs (k+2, k+3)
    int node = tile_m * 16 + row;       // A-side row this lane feeds
    int hcol = tile_n * 16 + row;       // B-side output column this lane feeds

    float inv = 1.0f / fmaxf(cnt[node], 1.0f);
    const float* aggRow = agg + (long)node * Fin;
    const float* xRow   = xin + (long)node * Fin;
    const float* WlRow  = Wl  + (long)hcol * Fin;
    const float* WrRow  = Wr  + (long)hcol * Fin;

    v8f c = {0.f, 0.f, 0.f, 0.f, 0.f, 0.f, 0.f, 0.f};
    for (int k = 0; k < Fin; k += 4) {
        int klo = k + (hi << 1);
        v2f a0 = { aggRow[klo] * inv, aggRow[klo + 1] * inv };
        v2f b0 = { WlRow[klo],        WlRow[klo + 1] };
        c = __builtin_amdgcn_wmma_f32_16x16x4_f32(false, a0, false, b0,
                                                  (short)0, c, false, false);
        v2f a1 = { xRow[klo],  xRow[klo + 1] };
        v2f b1 = { WrRow[klo], WrRow[klo + 1] };
        c = __builtin_amdgcn_wmma_f32_16x16x4_f32(false, a1, false, b1,
                                                  (short)0, c, false, false);
    }

    float bv = bias[hcol];
    int   col = tile_n * 16 + (lane & 15);
    #pragma unroll
    for (int v = 0; v < 8; ++v) {
        int   m   = v + (hi << 3);
        float val = c[v] + bv;
        if (do_relu) val = fmaxf(val, 0.0f);
        out[(long)(tile_m * 16 + m) * Hout + col] = val;
    }
}

// ---------------------------------------------------------------------------
// In-place log_softmax over 64 classes: one wave per row, 2 values per lane.
// ---------------------------------------------------------------------------
__global__ void log_softmax64(float* __restrict__ out, long n_rows) {
    long wave = ((long)blockIdx.x * blockDim.x + threadIdx.x) >> 5;
    int  lane = threadIdx.x & 31;
    if (wave >= n_rows) return;
    float* row = out + wave * 64;
    float v0 = row[lane];
    float v1 = row[lane + 32];
    float m = fmaxf(v0, v1);
    #pragma unroll
    for (int off = 16; off > 0; off >>= 1)
        m = fmaxf(m, __shfl_xor(m, off, 32));
    float s = expf(v0 - m) + expf(v1 - m);
    #pragma unroll
    for (int off = 16; off > 0; off >>= 1)
        s += __shfl_xor(s, off, 32);
    float lg = logf(s);
    row[lane]      = v0 - m - lg;
    row[lane + 32] = v1 - m - lg;
}

// ---------------------------------------------------------------------------
// Launcher: 3 x (zero, scatter, wmma-gemm) then log_softmax. All on stream.
// Workspace: agg[N*128] | cnt[N] | h1[N*128] | h2[N*128]  (~154 MB)
// ---------------------------------------------------------------------------
extern "C" void kernel_launch(void* const* d_in, const int* in_sizes, int n_in,
                              void* d_out, int out_size, void* d_ws, size_t ws_size,
                              hipStream_t stream) {
    const float* x   = (const float*)d_in[0];
    const int*   ei  = (const int*)d_in[1];
    const float* W0l = (const float*)d_in[2];
    const float* b0  = (const float*)d_in[3];
    const float* W0r = (const float*)d_in[4];
    const float* W1l = (const float*)d_in[5];
    const float* b1  = (const float*)d_in[6];
    const float* W1r = (const float*)d_in[7];
    const float* W2l = (const float*)d_in[8];
    const float* b2  = (const float*)d_in[9];
    const float* W2r = (const float*)d_in[10];

    const long N = (long)in_sizes[0] / 128;
    const long E = (long)in_sizes[1] / 2;

    float* agg = (float*)d_ws;
    float* cnt = agg + N * 128;
    float* h1  = cnt + N;
    float* h2  = h1 + N * 128;

    const long zero_n4 = (N * 129 + 3) / 4;           // agg + cnt contiguous
    const long zb = (zero_n4 + 255) / 256;
    const long sb = (E * 32 + 255) / 256;             // scatter blocks

    auto layer = [&](const float* xin, const float* Wl, const float* Wr,
                     const float* bias, float* out, int Hout, int do_relu) {
        fill_zero_f4<<<dim3((unsigned)zb), dim3(256), 0, stream>>>(agg, zero_n4);
        sage_scatter<<<dim3((unsigned)sb), dim3(256), 0, stream>>>(xin, ei, agg, cnt, E);
        long tiles = (N / 16) * (Hout / 16);
        long gb    = (tiles + 7) / 8;                 // 8 waves per 256-thread block
        sage_gemm_wmma<<<dim3((unsigned)gb), dim3(256), 0, stream>>>(
            agg, cnt, xin, Wl, Wr, bias, out, (int)N, 128, Hout, do_relu);
    };

    layer(x,  W0l, W0r, b0, h1, 128, 1);
    layer(h1, W1l, W1r, b1, h2, 128, 1);
    layer(h2, W2l, W2r, b2, (float*)d_out, 64, 0);

    const long lsb = (N + 7) / 8;                     // 8 waves per block, 1 row/wave
    log_softmax64<<<dim3((unsigned)lsb), dim3(256), 0, stream>>>((float*)d_out, N);
}